// TwoLayerGNNCls_70952859730187
// MI455X (gfx1250) — compile-verified
//
#include <hip/hip_runtime.h>
#include <hip/hip_bf16.h>

typedef __attribute__((ext_vector_type(2))) float v2f;
typedef __attribute__((ext_vector_type(8))) float v8f;

#define H_DIM 128
#define LDS_STRIDE 132            // 128 + 4 pad words -> conflict-free ds reads
#define ROWS_PER_BLOCK 64

// ---------------------------------------------------------------------------
// float4 copy: z = h  (initializes GIN aggregation with the self term)
// ---------------------------------------------------------------------------
__global__ __launch_bounds__(256)
void gin_copy_f4(const float* __restrict__ src, float* __restrict__ dst, int n4) {
    int i = blockIdx.x * 256 + threadIdx.x;
    if (i < n4) {
        ((float4*)dst)[i] = ((const float4*)src)[i];
    }
}

// ---------------------------------------------------------------------------
// Edge scatter-add: z[dst[e]] += h[src[e]]   (one wave per edge, float4/lane)
// ---------------------------------------------------------------------------
__global__ __launch_bounds__(256)
void gin_scatter_add(const float* __restrict__ h,
                     const int* __restrict__ src,
                     const int* __restrict__ dst,
                     float* __restrict__ z, int E) {
    int t = blockIdx.x * 256 + threadIdx.x;
    int e = t >> 5;
    if (e >= E) return;
    int off = (t & 31) << 2;                       // 0,4,...,124
    int s = __ldg(src + e);
    int d = __ldg(dst + e);
    float4 v = *(const float4*)(h + (size_t)s * H_DIM + off);
    float* p = z + (size_t)d * H_DIM + off;
    atomicAdd(p + 0, v.x);
    atomicAdd(p + 1, v.y);
    atomicAdd(p + 2, v.z);
    atomicAdd(p + 3, v.w);
}

// ---------------------------------------------------------------------------
// Pre-pack W [128x128] row-major into per-lane WMMA-B fragment order so the
// GEMM inner loop does one coalesced b128 load per lane per two K-steps.
// Layout: float4 index u = ((t*8 + wave)*32 + lane), t = K-pair index 0..15.
//   .xy = B pair for K-step 2t   (k = 8t   + klo)
//   .zw = B pair for K-step 2t+1 (k = 8t+4 + klo)
// with col = wave*16 + (lane&15), klo = 2*(lane>>4).
// ---------------------------------------------------------------------------
__global__ __launch_bounds__(256)
void gin_pack_w(const float* __restrict__ W, float* __restrict__ Wp) {
    int u = blockIdx.x * 256 + threadIdx.x;        // 4096 float4s
    if (u >= 4096) return;
    int lane = u & 31;
    int wave = (u >> 5) & 7;
    int t    = u >> 8;
    int ln   = lane & 15;
    int klo  = (lane >> 4) << 1;
    int col  = (wave << 4) + ln;
    int k0   = 8 * t + klo;
    int k1   = k0 + 4;
    float4 v;
    v.x = W[(k0 + 0) * H_DIM + col];
    v.y = W[(k0 + 1) * H_DIM + col];
    v.z = W[(k1 + 0) * H_DIM + col];
    v.w = W[(k1 + 1) * H_DIM + col];
    ((float4*)Wp)[u] = v;
}

// ---------------------------------------------------------------------------
// out = relu(A @ W + bias) with V_WMMA_F32_16X16X4_F32.
//   A   : [M,128] activations, Wp : packed weights, out : [M,128]
// Block = 256 threads = 8 waves, computes a 64-row x 128-col stripe.
// Wave w owns cols [16w,16w+16) across 4 row sub-tiles (B-frag reuse x4).
// ---------------------------------------------------------------------------
__global__ __launch_bounds__(256)
void gin_mlp_gemm(const float* __restrict__ A, const float* __restrict__ Wp,
                  const float* __restrict__ bias, float* __restrict__ out,
                  int M) {
    __shared__ float As[ROWS_PER_BLOCK * LDS_STRIDE];   // ~33 KB

    const int tid     = threadIdx.x;
    const int rowbase = blockIdx.x * ROWS_PER_BLOCK;

    // Cooperative load 64x128 activation tile (row-clamped for the tail block).
    for (int i = tid; i < ROWS_PER_BLOCK * 32; i += 256) {
        int r = i >> 5, c = i & 31;
        int gr = rowbase + r;
        if (gr >= M) gr = M - 1;
        ((float4*)As)[r * (LDS_STRIDE / 4) + c] =
            ((const float4*)A)[(size_t)gr * 32 + c];
    }
    __syncthreads();

    const int wave = tid >> 5;
    const int lane = tid & 31;
    const int colbase = wave << 4;
    const int half = lane >> 4;
    const int ln   = lane & 15;
    const int klo  = half << 1;

    v8f acc[4];
    #pragma unroll
    for (int rt = 0; rt < 4; ++rt)
        acc[rt] = (v8f){0.f, 0.f, 0.f, 0.f, 0.f, 0.f, 0.f, 0.f};

    const float4* wp = (const float4*)Wp + ((wave << 5) + lane);

    #pragma unroll 4
    for (int t = 0; t < 16; ++t) {
        float4 b4 = wp[t * 256];                   // coalesced b128, L2 hit
        v2f b0 = {b4.x, b4.y};                     // K-step 2t
        v2f b1 = {b4.z, b4.w};                     // K-step 2t+1
        const int ka = 8 * t + klo;
        #pragma unroll
        for (int rt = 0; rt < 4; ++rt) {
            const float* ap = &As[((rt << 4) + ln) * LDS_STRIDE + ka];
            v2f a0 = {ap[0], ap[1]};
            v2f a1 = {ap[4], ap[5]};
            acc[rt] = __builtin_amdgcn_wmma_f32_16x16x4_f32(
                false, a0, false, b0, (short)0, acc[rt], false, false);
            acc[rt] = __builtin_amdgcn_wmma_f32_16x16x4_f32(
                false, a1, false, b1, (short)0, acc[rt], false, false);
        }
    }

    // Epilogue: VGPR r of tile rt -> row rowbase + rt*16 + r + 8*half.
    const float bcol = bias[colbase + ln];
    #pragma unroll
    for (int rt = 0; rt < 4; ++rt) {
        #pragma unroll
        for (int r = 0; r < 8; ++r) {
            int row = rowbase + (rt << 4) + r + (half << 3);
            if (row < M) {
                float v = acc[rt][r] + bcol;
                v = v > 0.f ? v : 0.f;
                out[(size_t)row * H_DIM + colbase + ln] = v;
            }
        }
    }
}

// ---------------------------------------------------------------------------
// Zero small pooled buffers
// ---------------------------------------------------------------------------
__global__ void gin_zero(float* p, int n) {
    int i = blockIdx.x * 256 + threadIdx.x;
    if (i < n) p[i] = 0.f;
}

// ---------------------------------------------------------------------------
// Graph pooling: sums[g] += h[node], cnts[g] += 1
// ---------------------------------------------------------------------------
__global__ __launch_bounds__(256)
void gin_pool(const float* __restrict__ h, const int* __restrict__ gid,
              float* __restrict__ sums, float* __restrict__ cnts, int Nn) {
    int t = blockIdx.x * 256 + threadIdx.x;
    int node = t >> 5;
    if (node >= Nn) return;
    int off = (t & 31) << 2;
    int g = __ldg(gid + node);
    float4 v = *(const float4*)(h + (size_t)node * H_DIM + off);
    float* p = sums + g * H_DIM + off;
    atomicAdd(p + 0, v.x);
    atomicAdd(p + 1, v.y);
    atomicAdd(p + 2, v.z);
    atomicAdd(p + 3, v.w);
    if (off == 0) atomicAdd(cnts + g, 1.0f);
}

// ---------------------------------------------------------------------------
// Classifier: mean -> (Wc1, bc1, relu) -> (Wc2, bc2). One block, 16 graphs.
// ---------------------------------------------------------------------------
__global__ __launch_bounds__(256)
void gin_classifier(const float* __restrict__ sums, const float* __restrict__ cnts,
                    const float* __restrict__ Wc1, const float* __restrict__ bc1,
                    const float* __restrict__ Wc2, const float* __restrict__ bc2,
                    float* __restrict__ out, int G, int C) {
    __shared__ float hg[16 * H_DIM];
    __shared__ float t1[16 * H_DIM];
    int tid = threadIdx.x;
    for (int i = tid; i < G * H_DIM; i += 256) {
        int g = i >> 7;
        float c = cnts[g];
        c = c > 1.f ? c : 1.f;
        hg[i] = sums[i] / c;
    }
    __syncthreads();
    for (int i = tid; i < G * H_DIM; i += 256) {
        int g = i >> 7, col = i & (H_DIM - 1);
        float acc = bc1[col];
        for (int k = 0; k < H_DIM; ++k) acc += hg[g * H_DIM + k] * Wc1[k * H_DIM + col];
        t1[i] = acc > 0.f ? acc : 0.f;
    }
    __syncthreads();
    if (tid < G * C) {
        int g = tid / C, c = tid % C;
        float acc = bc2[c];
        for (int k = 0; k < H_DIM; ++k) acc += t1[g * H_DIM + k] * Wc2[k * C + c];
        out[g * C + c] = acc;
    }
}

// ---------------------------------------------------------------------------
// Launcher
// ---------------------------------------------------------------------------
extern "C" void kernel_launch(void* const* d_in, const int* in_sizes, int n_in,
                              void* d_out, int out_size, void* d_ws, size_t ws_size,
                              hipStream_t stream) {
    const float* h    = (const float*)d_in[0];
    const int*   src  = (const int*)d_in[1];
    const int*   dst  = (const int*)d_in[2];
    const int*   gid  = (const int*)d_in[3];
    const float* W1a  = (const float*)d_in[4];
    const float* b1a  = (const float*)d_in[5];
    const float* W2a  = (const float*)d_in[6];
    const float* b2a  = (const float*)d_in[7];
    const float* W1b  = (const float*)d_in[8];
    const float* b1b  = (const float*)d_in[9];
    const float* W2b  = (const float*)d_in[10];
    const float* b2b  = (const float*)d_in[11];
    const float* Wc1  = (const float*)d_in[12];
    const float* bc1  = (const float*)d_in[13];
    const float* Wc2  = (const float*)d_in[14];
    const float* bc2  = (const float*)d_in[15];

    const int N = in_sizes[3];         // graph_ids length
    const int E = in_sizes[1];         // src length
    const int G = 16;
    const int C = out_size / G;        // 10

    const size_t NH = (size_t)N * H_DIM;
    const int    WP = 16384;           // packed weight floats (64 KB)

    float* bufA  = (float*)d_ws;                   // z scratch
    float* bufB  = bufA + NH;                      // MLP mid
    float* bufC  = bufB + NH;                      // layer output
    float* sums  = bufC + NH;                      // [G, H]
    float* cnts  = sums + G * H_DIM;               // [G]
    float* Wp1a  = cnts + G;
    float* Wp2a  = Wp1a + WP;
    float* Wp1b  = Wp2a + WP;
    float* Wp2b  = Wp1b + WP;

    const int n4       = N * H_DIM / 4;
    const int copyGrid = (n4 + 255) / 256;
    const int edgeGrid = (E * 32 + 255) / 256;
    const int gemmGrid = (N + ROWS_PER_BLOCK - 1) / ROWS_PER_BLOCK;
    const int poolGrid = (N * 32 + 255) / 256;

    // ---- Pack weights into WMMA-B fragment order (tiny, once per launch) ----
    gin_pack_w<<<16, 256, 0, stream>>>(W1a, Wp1a);
    gin_pack_w<<<16, 256, 0, stream>>>(W2a, Wp2a);
    gin_pack_w<<<16, 256, 0, stream>>>(W1b, Wp1b);
    gin_pack_w<<<16, 256, 0, stream>>>(W2b, Wp2b);

    // ---- GIN layer A ----
    gin_copy_f4    <<<copyGrid, 256, 0, stream>>>(h, bufA, n4);
    gin_scatter_add<<<edgeGrid, 256, 0, stream>>>(h, src, dst, bufA, E);
    gin_mlp_gemm   <<<gemmGrid, 256, 0, stream>>>(bufA, Wp1a, b1a, bufB, N);
    gin_mlp_gemm   <<<gemmGrid, 256, 0, stream>>>(bufB, Wp2a, b2a, bufC, N);

    // ---- GIN layer B ----
    gin_copy_f4    <<<copyGrid, 256, 0, stream>>>(bufC, bufA, n4);
    gin_scatter_add<<<edgeGrid, 256, 0, stream>>>(bufC, src, dst, bufA, E);
    gin_mlp_gemm   <<<gemmGrid, 256, 0, stream>>>(bufA, Wp1b, b1b, bufB, N);
    gin_mlp_gemm   <<<gemmGrid, 256, 0, stream>>>(bufB, Wp2b, b2b, bufC, N);

    // ---- Pool + classify ----
    gin_zero      <<<(G * H_DIM + G + 255) / 256, 256, 0, stream>>>(sums, G * H_DIM + G);
    gin_pool      <<<poolGrid, 256, 0, stream>>>(bufC, gid, sums, cnts, N);
    gin_classifier<<<1, 256, 0, stream>>>(sums, cnts, Wc1, bc1, Wc2, bc2,
                                          (float*)d_out, G, C);
}